// SigmaCoordinateStepper_64141041598555
// MI455X (gfx1250) — compile-verified
//
#include <hip/hip_runtime.h>
#include <cstdint>
#include <cstddef>

// ---------------- problem constants ----------------
#define NLAT 361
#define NLON 720
#define KLEV 16
#define TI 8          // interior tile rows (lat)
#define TJ 16         // interior tile cols (lon)
#define HTI (TI + 2)  // +halo
#define HTJ (TJ + 2)

static constexpr float PI_F     = 3.14159265358979323846f;
static constexpr float R_GAS    = 287.0f;
static constexpr float C_P      = 1004.0f;
static constexpr float R_EARTH  = 6371000.0f;
static constexpr float OMEGA_E  = 7.292e-05f;
static constexpr float LSRC     = 0.10729586082894002f;   // ln(5)/15 = ln(sig_k/sig_{k+1}), constant
static constexpr float DTH      = PI_F / (NLAT - 1);      // = pi/360
static constexpr float DLAM     = 2.0f * PI_F / NLON;     // = pi/360
static constexpr float INV_DTH  = 1.0f / DTH;
static constexpr float INV2DLAM = 0.5f / DLAM;
static constexpr float INV_RE   = 1.0f / R_EARTH;

// ---------------- CDNA5 async global->LDS path (guarded) ----------------
#if defined(__AMDGCN__) && __has_builtin(__builtin_amdgcn_global_load_async_to_lds_b32) && \
    __has_builtin(__builtin_amdgcn_global_load_async_to_lds_b64)
#define ASYNC_LDS 1
#else
#define ASYNC_LDS 0
#endif

#if ASYNC_LDS
typedef int v2i_t __attribute__((ext_vector_type(2)));
typedef __attribute__((address_space(1))) int   as1_i32;
typedef __attribute__((address_space(3))) int   as3_i32;
typedef __attribute__((address_space(1))) v2i_t as1_v2i;
typedef __attribute__((address_space(3))) v2i_t as3_v2i;
#endif

// AS1 pointers are 64-bit; AS3 pointers are 32-bit LDS offsets. Per the ISA
// flat-aperture mapping, the LDS address is the low 32 bits of the generic one, so truncating the
// generic address of a __shared__ object gives the correct AS3 pointer value.
__device__ __forceinline__ void cp_b32(const float* __restrict__ g, float* l) {
#if ASYNC_LDS
  // GLOBAL_LOAD_ASYNC_TO_LDS_B32: copies 4 bytes from the global address in g to the LDS address in l, tracked by ASYNCcnt
  __builtin_amdgcn_global_load_async_to_lds_b32(
      (as1_i32*)(uintptr_t)g, (as3_i32*)(uint32_t)(uintptr_t)l, 0, 0);
#else
  *l = *g;
#endif
}

__device__ __forceinline__ void cp_b64(const float* __restrict__ g, float* l) {
#if ASYNC_LDS
  __builtin_amdgcn_global_load_async_to_lds_b64(
      (as1_v2i*)(uintptr_t)g, (as3_v2i*)(uint32_t)(uintptr_t)l, 0, 0);
#else
  l[0] = g[0];
  l[1] = g[1];
#endif
}

__device__ __forceinline__ void wait_async_loads() {
#if ASYNC_LDS
#if __has_builtin(__builtin_amdgcn_s_wait_asynccnt)
  __builtin_amdgcn_s_wait_asynccnt(0);
#else
  asm volatile("s_wait_asynccnt 0" ::: "memory");
#endif
#endif
}

// ---------------- non-temporal output stores (outputs > L2, stream them) ----------------
typedef __attribute__((ext_vector_type(2))) float f32x2;

__device__ __forceinline__ void st_nt(float* p, float v) {
  __builtin_nontemporal_store(v, p);
}
__device__ __forceinline__ void st_nt2(float* p, float a, float b) {
  f32x2 v;
  v.x = a;
  v.y = b;
  __builtin_nontemporal_store(v, (f32x2*)p);
}

// one-sided at poles, centered interior (matches reference _d_dth)
__device__ __forceinline__ float ddth(int i, float xm, float xc, float xp) {
  if (i == 0) return (xp - xc) * INV_DTH;
  if (i == NLAT - 1) return (xc - xm) * INV_DTH;
  return (xp - xm) * (0.5f * INV_DTH);
}

__device__ __forceinline__ int iclampi(int v, int lo, int hi) {
  return v < lo ? lo : (v > hi ? hi : v);
}

__global__ __launch_bounds__(TI * TJ) void sigma_step_kernel(
    const float* __restrict__ uv,   // (B,K,H,W,2)
    const float* __restrict__ Tg,   // (B,K,H,W)
    const float* __restrict__ qg,   // (B,K,H,W)
    const float* __restrict__ psg,  // (B,H,W)
    float* __restrict__ o_duv,      // (B,K,H,W,2)
    float* __restrict__ o_dT,       // (B,K,H,W)
    float* __restrict__ o_dq,       // (B,K,H,W)
    float* __restrict__ o_dps)      // (B,H,W)
{
  // LDS staging tile: 47,520 bytes
  __shared__ __align__(16) float s_uv[KLEV][HTI][HTJ][2];
  __shared__ float s_T[KLEV][HTI][HTJ];
  __shared__ float s_q[KLEV][HTI][HTJ];
  __shared__ float s_ps[HTI][HTJ];
  __shared__ float s_lps[HTI][HTJ];

  const int b  = blockIdx.z;
  const int i0 = blockIdx.y * TI;
  const int j0 = blockIdx.x * TJ;
  const int tx = threadIdx.x;  // 0..15 (lon)
  const int ty = threadIdx.y;  // 0..7  (lat)
  const int tid = ty * TJ + tx;

  // ---- stage tile (center + 1-halo) into LDS with async-to-LDS ops ----
  for (int c = tid; c < HTI * HTJ; c += TI * TJ) {
    const int ii = c / HTJ, jj = c % HTJ;
    const int gi = iclampi(i0 - 1 + ii, 0, NLAT - 1);
    const int gj = (j0 - 1 + jj + NLON) % NLON;
    cp_b32(psg + ((size_t)b * NLAT + gi) * NLON + gj, &s_ps[ii][jj]);
  }
  for (int t = tid; t < HTI * KLEV * HTJ; t += TI * TJ) {
    const int jj = t % HTJ;
    const int r  = t / HTJ;
    const int k  = r % KLEV;
    const int ii = r / KLEV;
    const int gi = iclampi(i0 - 1 + ii, 0, NLAT - 1);
    const int gj = (j0 - 1 + jj + NLON) % NLON;
    const size_t base = (((size_t)(b * KLEV + k) * NLAT + gi) * NLON + gj);
    cp_b64(uv + base * 2, &s_uv[k][ii][jj][0]);  // both velocity comps in one B64
    cp_b32(Tg + base, &s_T[k][ii][jj]);
    cp_b32(qg + base, &s_q[k][ii][jj]);
  }
  wait_async_loads();
  __syncthreads();

  // log(p_s) tile
  for (int c = tid; c < HTI * HTJ; c += TI * TJ) {
    const int ii = c / HTJ, jj = c % HTJ;
    s_lps[ii][jj] = logf(s_ps[ii][jj]);
  }
  __syncthreads();

  const int i = i0 + ty;
  const int j = j0 + tx;
  if (i >= NLAT) return;  // no barriers below this point

  const int ii = ty + 1, jj = tx + 1;

  // per-row trig (halo rows hold clamped duplicates; the boundary branch of
  // ddth never reads the clamped sample, so sinm/sinp at poles are dont-cares)
  const float colat = DTH * (float)i;
  const float sinc  = fmaxf(sinf(colat), 1e-5f);
  const float sinm  = fmaxf(sinf(DTH * (float)(i - 1)), 1e-5f);
  const float sinp  = fmaxf(sinf(DTH * (float)(i + 1)), 1e-5f);
  const float fcor  = 2.0f * OMEGA_E * cosf(colat);  // sin(lat) = cos(colat)
  const float invRsin = INV_RE / sinc;

  // sigma levels (geometric) and layer thicknesses
  float sig[KLEV], dsig[KLEV];
#pragma unroll
  for (int k = 0; k < KLEV; ++k) sig[k] = expf(-LSRC * (float)k);
  dsig[0] = sig[0] - 0.5f * (sig[0] + sig[1]);
#pragma unroll
  for (int k = 1; k < KLEV - 1; ++k) dsig[k] = 0.5f * (sig[k - 1] - sig[k + 1]);
  dsig[KLEV - 1] = 0.5f * (sig[KLEV - 2] + sig[KLEV - 1]);
  const float invds_first = 1.0f / (sig[1] - sig[0]);
  const float invds_last  = 1.0f / (sig[KLEV - 1] - sig[KLEV - 2]);

  // grad(ln p_s) (level independent)
  const float g_t = ddth(i, s_lps[ii - 1][jj], s_lps[ii][jj], s_lps[ii + 1][jj]) * INV_RE;
  const float g_l = (s_lps[ii][jj + 1] - s_lps[ii][jj - 1]) * INV2DLAM * invRsin;

  // ---- sweep 1: flux_k = (div + V.grad ln p_s) * dsig_k ----
  float flux[KLEV];
  float total = 0.0f;
#pragma unroll
  for (int k = 0; k < KLEV; ++k) {
    const float vtC = s_uv[k][ii][jj][0];
    const float vlC = s_uv[k][ii][jj][1];
    const float vtN = s_uv[k][ii - 1][jj][0];
    const float vtS = s_uv[k][ii + 1][jj][0];
    const float vlW = s_uv[k][ii][jj - 1][1];
    const float vlE = s_uv[k][ii][jj + 1][1];
    const float div = (ddth(i, sinm * vtN, sinc * vtC, sinp * vtS) +
                       (vlE - vlW) * INV2DLAM) * invRsin;
    const float vgl = vtC * g_t + vlC * g_l;
    flux[k] = (div + vgl) * dsig[k];
    total += flux[k];
  }

  const float ps_c = s_ps[ii][jj];
  st_nt(o_dps + ((size_t)b * NLAT + i) * NLON + j, -ps_c * total);

  // ---- sweep 2: tendencies ----
  float csum = 0.0f;
  float phiC = 0.f, phiN = 0.f, phiS = 0.f, phiE = 0.f, phiW = 0.f;
  float pTc = 0.f, pTn = 0.f, pTs = 0.f, pTe = 0.f, pTw = 0.f;
  const float RLSR = R_GAS * LSRC;

#pragma unroll
  for (int k = 0; k < KLEV; ++k) {
    // hydrostatic phi at the 5 stencil points (cumsum of R*T*ln(sig ratio))
    if (k > 0) {
      phiC += RLSR * pTc;
      phiN += RLSR * pTn;
      phiS += RLSR * pTs;
      phiE += RLSR * pTe;
      phiW += RLSR * pTw;
    }

    const float Tc = s_T[k][ii][jj];
    const float Tn = s_T[k][ii - 1][jj];
    const float Ts = s_T[k][ii + 1][jj];
    const float Tw = s_T[k][ii][jj - 1];
    const float Te = s_T[k][ii][jj + 1];

    const float qc = s_q[k][ii][jj];
    const float qn = s_q[k][ii - 1][jj];
    const float qs = s_q[k][ii + 1][jj];
    const float qw = s_q[k][ii][jj - 1];
    const float qe = s_q[k][ii][jj + 1];

    const float vtC = s_uv[k][ii][jj][0], vlC = s_uv[k][ii][jj][1];
    const float vtN = s_uv[k][ii - 1][jj][0], vlN = s_uv[k][ii - 1][jj][1];
    const float vtS = s_uv[k][ii + 1][jj][0], vlS = s_uv[k][ii + 1][jj][1];
    const float vtW = s_uv[k][ii][jj - 1][0], vlW = s_uv[k][ii][jj - 1][1];
    const float vtE = s_uv[k][ii][jj + 1][0], vlE = s_uv[k][ii][jj + 1][1];

    // relative vorticity
    const float zeta = (ddth(i, sinm * vlN, sinc * vlC, sinp * vlS) -
                        (vtE - vtW) * INV2DLAM) * invRsin;

    // grad(phi + KE)
    const float sC = phiC + 0.5f * (vtC * vtC + vlC * vlC);
    const float sN = phiN + 0.5f * (vtN * vtN + vlN * vlN);
    const float sS = phiS + 0.5f * (vtS * vtS + vlS * vlS);
    const float sW = phiW + 0.5f * (vtW * vtW + vlW * vlW);
    const float sE = phiE + 0.5f * (vtE * vtE + vlE * vlE);
    const float gpk_t = ddth(i, sN, sC, sS) * INV_RE;
    const float gpk_l = (sE - sW) * INV2DLAM * invRsin;

    // vertical derivatives (nonuniform central, one-sided at ends)
    float dvt_ds, dvl_ds, dT_ds, dq_ds;
    if (k == 0) {
      dvt_ds = (s_uv[1][ii][jj][0] - vtC) * invds_first;
      dvl_ds = (s_uv[1][ii][jj][1] - vlC) * invds_first;
      dT_ds  = (s_T[1][ii][jj] - Tc) * invds_first;
      dq_ds  = (s_q[1][ii][jj] - qc) * invds_first;
    } else if (k == KLEV - 1) {
      dvt_ds = (vtC - s_uv[KLEV - 2][ii][jj][0]) * invds_last;
      dvl_ds = (vlC - s_uv[KLEV - 2][ii][jj][1]) * invds_last;
      dT_ds  = (Tc - s_T[KLEV - 2][ii][jj]) * invds_last;
      dq_ds  = (qc - s_q[KLEV - 2][ii][jj]) * invds_last;
    } else {
      const float invdd = 1.0f / (sig[k + 1] - sig[k - 1]);
      dvt_ds = (s_uv[k + 1][ii][jj][0] - s_uv[k - 1][ii][jj][0]) * invdd;
      dvl_ds = (s_uv[k + 1][ii][jj][1] - s_uv[k - 1][ii][jj][1]) * invdd;
      dT_ds  = (s_T[k + 1][ii][jj] - s_T[k - 1][ii][jj]) * invdd;
      dq_ds  = (s_q[k + 1][ii][jj] - s_q[k - 1][ii][jj]) * invdd;
    }

    // sigma_dot_k = 0.5*(sdh_k + sdh_{k+1}) = -(csum_{k-1} + 0.5*flux_k)
    const float sdot = -(csum + 0.5f * flux[k]);
    csum += flux[k];

    const float vgl = vtC * g_t + vlC * g_l;
    const float zf  = zeta + fcor;

    // momentum: -(zeta+f) k x V - grad(phi+KE) - sdot dV/ds - R T grad(ln p_s)
    const float duv_t = zf * vlC - gpk_t - sdot * dvt_ds - R_GAS * Tc * g_t;
    const float duv_l = -zf * vtC - gpk_l - sdot * dvl_ds - R_GAS * Tc * g_l;

    // thermodynamic
    const float oop  = -total + vgl + sdot / sig[k];
    const float gT_t = ddth(i, Tn, Tc, Ts) * INV_RE;
    const float gT_l = (Te - Tw) * INV2DLAM * invRsin;
    const float dT_dt =
        -(vtC * gT_t + vlC * gT_l) - sdot * dT_ds + (R_GAS / C_P) * Tc * oop;

    // tracer
    const float gq_t = ddth(i, qn, qc, qs) * INV_RE;
    const float gq_l = (qe - qw) * INV2DLAM * invRsin;
    const float dq_dt = -(vtC * gq_t + vlC * gq_l) - sdot * dq_ds;

    const size_t pidx = (((size_t)(b * KLEV + k) * NLAT + i) * NLON + j);
    st_nt2(o_duv + pidx * 2, duv_t, duv_l);
    st_nt(o_dT + pidx, dT_dt);
    st_nt(o_dq + pidx, dq_dt);

    pTc = Tc; pTn = Tn; pTs = Ts; pTe = Te; pTw = Tw;
  }
}

extern "C" void kernel_launch(void* const* d_in, const int* in_sizes, int n_in,
                              void* d_out, int out_size, void* d_ws, size_t ws_size,
                              hipStream_t stream) {
  const float* uv = (const float*)d_in[0];
  const float* T  = (const float*)d_in[1];
  const float* q  = (const float*)d_in[2];
  const float* ps = (const float*)d_in[3];

  const int B = in_sizes[3] / (NLAT * NLON);

  float* out = (float*)d_out;
  const size_t n3 = (size_t)B * KLEV * NLAT * NLON;
  float* o_duv = out;             // (B,K,H,W,2)
  float* o_dT  = out + n3 * 2;    // (B,K,H,W)
  float* o_dq  = o_dT + n3;       // (B,K,H,W)
  float* o_dps = o_dq + n3;       // (B,H,W)

  dim3 grid((NLON + TJ - 1) / TJ, (NLAT + TI - 1) / TI, B);
  dim3 block(TJ, TI);
  hipLaunchKernelGGL(sigma_step_kernel, grid, block, 0, stream,
                     uv, T, q, ps, o_duv, o_dT, o_dq, o_dps);
}